// GCN_8770323219097
// MI455X (gfx1250) — compile-verified
//
#include <hip/hip_runtime.h>
#include <hip/hip_bf16.h>

#define IN_DIM  256
#define HID_DIM 256

typedef __attribute__((ext_vector_type(2))) float v2f;
typedef __attribute__((ext_vector_type(8))) float v8f;

// ---------------- degree / norm ----------------
__global__ void k_set1(float* __restrict__ p, int n) {
    int i = blockIdx.x * blockDim.x + threadIdx.x;
    if (i < n) p[i] = 1.0f;                 // self-loop contributes 1 to degree
}

__global__ void k_deg(const int* __restrict__ ei, float* __restrict__ deg, int E) {
    int e = blockIdx.x * blockDim.x + threadIdx.x;
    if (e < E) atomicAdd(&deg[ei[E + e]], 1.0f);   // dst row of edge_index
}

__global__ void k_rsqrt(float* __restrict__ p, int n) {
    int i = blockIdx.x * blockDim.x + threadIdx.x;
    if (i < n) p[i] = rsqrtf(p[i]);         // deg >= 1 always
}

// ---------------- GEMM: C[M,256] = (relu?)A[M,256] @ W[256,256] ----------------
// block = 128 threads (4 waves). Block covers 64 rows x 256 cols.
// Each wave: 4 M-tiles x 4 N-tiles = 64x64 output, 16 v8f accumulators.
// Per K-step (K+=4): 4 A b64-loads + 8 B b32-loads feed 16 WMMAs (B reused across M-tiles,
// A reused across N-tiles).
// WMMA f32 16x16x4 fragment layout:
//   A: lane l<16 holds K=k0,k0+1 ; lane l>=16 holds K=k0+2,k0+3 (M = l&15).
//   B mirrors: v0 = W[k0+2*half][n], v1 = W[k0+1+2*half][n], n = n0+(l&15).
template <bool RELU>
__global__ __launch_bounds__(128) void k_gemm(const float* __restrict__ A,
                                              const float* __restrict__ W,
                                              float* __restrict__ C, int M) {
    const int lane = threadIdx.x & 31;
    const int wave = threadIdx.x >> 5;
    const int half = lane >> 4;      // 0|1
    const int l16  = lane & 15;
    const int m0   = blockIdx.x * 64;
    const int n0   = wave * 64;

    const float* arow[4];
#pragma unroll
    for (int mt = 0; mt < 4; ++mt) {
        int row  = m0 + mt * 16 + l16;
        int rowc = row < M ? row : (M - 1);          // clamp loads for tail block
        arow[mt] = A + (size_t)rowc * IN_DIM + 2 * half;
    }

    v8f acc[4][4];
#pragma unroll
    for (int mt = 0; mt < 4; ++mt)
#pragma unroll
        for (int nt = 0; nt < 4; ++nt) acc[mt][nt] = (v8f){};

    for (int k0 = 0; k0 < IN_DIM; k0 += 4) {
        v2f a[4];
#pragma unroll
        for (int mt = 0; mt < 4; ++mt) {
            float2 av = *(const float2*)(arow[mt] + k0);
            if (RELU) { av.x = fmaxf(av.x, 0.0f); av.y = fmaxf(av.y, 0.0f); }
            a[mt].x = av.x; a[mt].y = av.y;
        }

        const float* wb = W + (size_t)(k0 + 2 * half) * HID_DIM + n0 + l16;
        v2f b[4];
#pragma unroll
        for (int nt = 0; nt < 4; ++nt) {
            b[nt].x = wb[nt * 16];
            b[nt].y = wb[HID_DIM + nt * 16];
        }

#pragma unroll
        for (int mt = 0; mt < 4; ++mt)
#pragma unroll
            for (int nt = 0; nt < 4; ++nt)
                acc[mt][nt] = __builtin_amdgcn_wmma_f32_16x16x4_f32(
                    false, a[mt], false, b[nt], (short)0, acc[mt][nt], false, false);
    }

    // C/D layout: lane l<16 -> VGPR r = C[mrow0+r][n]; lane l>=16 -> C[mrow0+8+r][n]
#pragma unroll
    for (int mt = 0; mt < 4; ++mt) {
        const int crow0 = m0 + mt * 16 + half * 8;
        float* cbase = C + (size_t)crow0 * HID_DIM + n0 + l16;
#pragma unroll
        for (int r = 0; r < 8; ++r) {
            if (crow0 + r < M) {
                cbase[(size_t)r * HID_DIM +  0] = acc[mt][0][r];
                cbase[(size_t)r * HID_DIM + 16] = acc[mt][1][r];
                cbase[(size_t)r * HID_DIM + 32] = acc[mt][2][r];
                cbase[(size_t)r * HID_DIM + 48] = acc[mt][3][r];
            }
        }
    }
}

// ---------------- AGG init: bias + self-loop term ----------------
__global__ void k_init_agg(const float* __restrict__ H, const float* __restrict__ dis,
                           const float* __restrict__ b, float* __restrict__ AGG, int Nn) {
    int idx = blockIdx.x * blockDim.x + threadIdx.x;      // float4 granularity
    int total = Nn * (HID_DIM / 4);
    if (idx >= total) return;
    int v  = idx / (HID_DIM / 4);
    int c4 = idx - v * (HID_DIM / 4);
    float s = dis[v]; s = s * s;                          // self-loop norm = dis^2
    float4 h  = ((const float4*)H)[idx];
    float4 bb = ((const float4*)b)[c4];
    float4 r;
    r.x = bb.x + h.x * s; r.y = bb.y + h.y * s;
    r.z = bb.z + h.z * s; r.w = bb.w + h.w * s;
    ((float4*)AGG)[idx] = r;
}

// ---------------- edge scatter: one wave per edge, L2-resident atomics ----------------
__global__ __launch_bounds__(256) void k_scatter(const float* __restrict__ H,
                                                 const float* __restrict__ dis,
                                                 const int* __restrict__ ei,
                                                 float* __restrict__ AGG, int E) {
    int e = blockIdx.x * 8 + (threadIdx.x >> 5);
    if (e >= E) return;
    int lane = threadIdx.x & 31;
    int s = ei[e];
    int d = ei[E + e];
    float nm = dis[s] * dis[d];
    const float4* hs = (const float4*)(H + (size_t)s * HID_DIM);
    float* ad = AGG + (size_t)d * HID_DIM;
#pragma unroll
    for (int i = 0; i < 2; ++i) {
        int c4 = lane + i * 32;                 // 64 float4 per row
        float4 h = hs[c4];
        int c = c4 * 4;
        atomicAdd(ad + c + 0, h.x * nm);
        atomicAdd(ad + c + 1, h.y * nm);
        atomicAdd(ad + c + 2, h.z * nm);
        atomicAdd(ad + c + 3, h.w * nm);
    }
}

// ---------------- FC(256->2) + log_softmax: one wave per node ----------------
__global__ __launch_bounds__(256) void k_fc(const float* __restrict__ AGG,
                                            const float* __restrict__ Wfc,
                                            const float* __restrict__ bfc,
                                            float* __restrict__ out, int Nn) {
    int v = blockIdx.x * 8 + (threadIdx.x >> 5);
    if (v >= Nn) return;
    int lane = threadIdx.x & 31;
    const float* h = AGG + (size_t)v * HID_DIM;
    float a0 = 0.0f, a1 = 0.0f;
#pragma unroll
    for (int c = lane; c < HID_DIM; c += 32) {
        float x = fmaxf(h[c], 0.0f);            // ReLU fused here
        a0 += x * Wfc[c * 2 + 0];
        a1 += x * Wfc[c * 2 + 1];
    }
#pragma unroll
    for (int off = 16; off > 0; off >>= 1) {
        a0 += __shfl_xor(a0, off, 32);
        a1 += __shfl_xor(a1, off, 32);
    }
    if (lane == 0) {
        float l0 = a0 + bfc[0];
        float l1 = a1 + bfc[1];
        float m = fmaxf(l0, l1);
        float lse = m + logf(expf(l0 - m) + expf(l1 - m));
        out[(size_t)v * 2 + 0] = l0 - lse;
        out[(size_t)v * 2 + 1] = l1 - lse;
    }
}

extern "C" void kernel_launch(void* const* d_in, const int* in_sizes, int n_in,
                              void* d_out, int out_size, void* d_ws, size_t ws_size,
                              hipStream_t stream) {
    const float* x   = (const float*)d_in[0];
    const int*   ei  = (const int*)d_in[1];   // [2, E]
    const float* W1  = (const float*)d_in[2];
    const float* b1  = (const float*)d_in[3];
    const float* W2  = (const float*)d_in[4];
    const float* b2  = (const float*)d_in[5];
    const float* Wfc = (const float*)d_in[6];
    const float* bfc = (const float*)d_in[7];
    float* out = (float*)d_out;

    const int Nn = in_sizes[0] / IN_DIM;   // 50000
    const int E  = in_sizes[1] / 2;        // 800000

    // workspace layout: dis | H | AGG   (~103 MB; H and AGG both fit in 192MB L2)
    char* ws = (char*)d_ws;
    float* dis = (float*)ws;
    size_t off = (((size_t)Nn * 4) + 255) & ~(size_t)255;
    float* H   = (float*)(ws + off);
    float* AGG = (float*)(ws + off + (size_t)Nn * HID_DIM * sizeof(float));

    const int T = 256;
    // degrees -> dis = rsqrt(deg)
    k_set1 <<<(Nn + T - 1) / T, T, 0, stream>>>(dis, Nn);
    k_deg  <<<(E  + T - 1) / T, T, 0, stream>>>(ei, dis, E);
    k_rsqrt<<<(Nn + T - 1) / T, T, 0, stream>>>(dis, Nn);

    const int mtiles = (Nn + 63) / 64;
    const int t4 = Nn * (HID_DIM / 4);

    // layer 1
    k_gemm<false><<<mtiles, 128, 0, stream>>>(x, W1, H, Nn);
    k_init_agg<<<(t4 + T - 1) / T, T, 0, stream>>>(H, dis, b1, AGG, Nn);
    k_scatter <<<(E + 7) / 8, T, 0, stream>>>(H, dis, ei, AGG, E);

    // layer 2 (ReLU fused into GEMM A-load)
    k_gemm<true><<<mtiles, 128, 0, stream>>>(AGG, W2, H, Nn);
    k_init_agg<<<(t4 + T - 1) / T, T, 0, stream>>>(H, dis, b2, AGG, Nn);
    k_scatter <<<(E + 7) / 8, T, 0, stream>>>(H, dis, ei, AGG, E);

    // FC + log_softmax (ReLU fused into load)
    k_fc<<<(Nn + 7) / 8, T, 0, stream>>>(AGG, Wfc, bfc, out, Nn);
}